// PCFLoss_8272107012336
// MI455X (gfx1250) — compile-verified
//
#include <hip/hip_runtime.h>
#include <math.h>

#define BB   4
#define NN   2048
#define NBIN 10
#define PP   5
#define BIGV 100000.0f
#define BN   (BB*NN)
#define NT16 (NN/16)

typedef __attribute__((ext_vector_type(2))) float v2f;
typedef __attribute__((ext_vector_type(8))) float v8f;

__device__ __forceinline__ float softplusf(float x) {
    return fmaxf(x, 0.0f) + log1pf(expf(-fabsf(x)));
}

// Select the (k, k+1) pair for this lane-half from a float4 covering k=4j..4j+3.
// A-matrix f32 16x16x4 layout: lanes 0-15 hold {K=4j, 4j+1}, lanes 16-31 hold {K=4j+2, 4j+3}.
__device__ __forceinline__ v2f pick2(float4 v, int grp) {
    v2f r;
    r[0] = grp ? v.z : v.x;
    r[1] = grp ? v.w : v.y;
    return r;
}

// Chain of four K=4 fp32 WMMAs => full K=16 dot-product tile (16x16 f32 accum).
__device__ __forceinline__ v8f wmma_k16(v2f a0, v2f a1, v2f a2, v2f a3,
                                        v2f b0, v2f b1, v2f b2, v2f b3) {
    v8f acc = {0.f,0.f,0.f,0.f,0.f,0.f,0.f,0.f};
    acc = __builtin_amdgcn_wmma_f32_16x16x4_f32(false, a0, false, b0, (short)0, acc, false, false);
    acc = __builtin_amdgcn_wmma_f32_16x16x4_f32(false, a1, false, b1, (short)0, acc, false, false);
    acc = __builtin_amdgcn_wmma_f32_16x16x4_f32(false, a2, false, b2, (short)0, acc, false, false);
    acc = __builtin_amdgcn_wmma_f32_16x16x4_f32(false, a3, false, b3, (short)0, acc, false, false);
    return acc;
}

// ---------------------------------------------------------------------------
// Kernel 1: per-(b,n) grasp construction, control-point transforms, norms,
// and elementwise loss terms.
// ---------------------------------------------------------------------------
__global__ void pcf_setup(
    const float* __restrict__ gdir, const float* __restrict__ adir,
    const float* __restrict__ goff, const float* __restrict__ ppts,
    const float* __restrict__ bsp,  const float* __restrict__ bsh,
    const float* __restrict__ dlab, const float* __restrict__ olab,
    const float* __restrict__ succ, const float* __restrict__ alab,
    const float* __restrict__ binv, const float* __restrict__ binw,
    const float* __restrict__ cpts, const float* __restrict__ scpts,
    float* __restrict__ Apred, float* __restrict__ Gpos, float* __restrict__ Gsym,
    float* __restrict__ Ggt,   float* __restrict__ Gsgt,
    float* __restrict__ n_pred, float* __restrict__ n_pos, float* __restrict__ n_sym,
    float* __restrict__ n_gt,   float* __restrict__ n_sgt,
    float* __restrict__ cosd_t, float* __restrict__ cosa_t,
    float* __restrict__ off_t,  float* __restrict__ bce_t)
{
    int i = blockIdx.x * blockDim.x + threadIdx.x;
    if (i >= BN) return;

    // predicted frame: base = grasp_dir, approach = approach_dir, cross = approach x base
    float bx = gdir[i*3+0], by = gdir[i*3+1], bz = gdir[i*3+2];
    float ax = adir[i*3+0], ay = adir[i*3+1], az = adir[i*3+2];
    float crx = ay*bz - az*by;
    float cry = az*bx - ax*bz;
    float crz = ax*by - ay*bx;

    // thickness via argmax (first-max semantics)
    float mv = goff[i*NBIN]; int mi = 0;
    #pragma unroll
    for (int k = 1; k < NBIN; ++k) { float v = goff[i*NBIN+k]; if (v > mv) { mv = v; mi = k; } }
    float thp = binv[mi];
    mv = olab[i*NBIN]; mi = 0;
    #pragma unroll
    for (int k = 1; k < NBIN; ++k) { float v = olab[i*NBIN+k]; if (v > mv) { mv = v; mi = k; } }
    float thg = binv[mi];

    float px = ppts[i*3+0], py = ppts[i*3+1], pz = ppts[i*3+2];
    float tx = px + 0.5f*thp*bx, ty = py + 0.5f*thp*by, tz = pz + 0.5f*thp*bz;

    // GT frame
    float lbx = dlab[i*3+0], lby = dlab[i*3+1], lbz = dlab[i*3+2];
    float lax = alab[i*3+0], lay = alab[i*3+1], laz = alab[i*3+2];
    float lcx = lay*lbz - laz*lby;
    float lcy = laz*lbx - lax*lbz;
    float lcz = lax*lby - lay*lbx;
    float tgx = px + 0.5f*thg*lbx, tgy = py + 0.5f*thg*lby, tgz = pz + 0.5f*thg*lbz;

    float s = succ[i];
    bool posf = (s != 0.0f);

    float npred=0.f, npos=0.f, nsym=0.f, ngt=0.f, nsgt=0.f;
    size_t base16 = (size_t)i * 16;

    #pragma unroll
    for (int p = 0; p < PP; ++p) {
        float cx = cpts[p*3+0], cy = cpts[p*3+1], cz = cpts[p*3+2];
        float sx = scpts[p*3+0], sy = scpts[p*3+1], sz = scpts[p*3+2];

        // predicted control point
        float qx = bx*cx + crx*cy + ax*cz + tx;
        float qy = by*cx + cry*cy + ay*cz + ty;
        float qz = bz*cx + crz*cy + az*cz + tz;
        Apred[base16+p*3+0]=qx; Apred[base16+p*3+1]=qy; Apred[base16+p*3+2]=qz;
        npred += qx*qx + qy*qy + qz*qz;

        // GT control point (unmasked)
        float gx = lbx*cx + lcx*cy + lax*cz + tgx;
        float gy = lby*cx + lcy*cy + lay*cz + tgy;
        float gz = lbz*cx + lcz*cy + laz*cz + tgz;
        Ggt[base16+p*3+0]=gx; Ggt[base16+p*3+1]=gy; Ggt[base16+p*3+2]=gz;
        ngt += gx*gx + gy*gy + gz*gz;

        // sym GT control point (unmasked)
        float hx = lbx*sx + lcx*sy + lax*sz + tgx;
        float hy = lby*sx + lcy*sy + lay*sz + tgy;
        float hz = lbz*sx + lcz*sy + laz*sz + tgz;
        Gsgt[base16+p*3+0]=hx; Gsgt[base16+p*3+1]=hy; Gsgt[base16+p*3+2]=hz;
        nsgt += hx*hx + hy*hy + hz*hz;

        // masked (success-gated) versions: whole 4x4 replaced by BIG when succ==0
        float mgx=gx, mgy=gy, mgz=gz, mhx=hx, mhy=hy, mhz=hz;
        if (!posf) {
            float vb = BIGV * (cx + cy + cz + 1.0f);
            mgx = mgy = mgz = vb;
            float vs = BIGV * (sx + sy + sz + 1.0f);
            mhx = mhy = mhz = vs;
        }
        Gpos[base16+p*3+0]=mgx; Gpos[base16+p*3+1]=mgy; Gpos[base16+p*3+2]=mgz;
        npos += mgx*mgx + mgy*mgy + mgz*mgz;
        Gsym[base16+p*3+0]=mhx; Gsym[base16+p*3+1]=mhy; Gsym[base16+p*3+2]=mhz;
        nsym += mhx*mhx + mhy*mhy + mhz*mhz;
    }
    // K padding slot = 0
    Apred[base16+15]=0.f; Ggt[base16+15]=0.f; Gsgt[base16+15]=0.f;
    Gpos[base16+15]=0.f;  Gsym[base16+15]=0.f;
    n_pred[i]=npred; n_pos[i]=npos; n_sym[i]=nsym; n_gt[i]=ngt; n_sgt[i]=nsgt;

    // elementwise losses
    float cosd = 1.0f - (lbx*bx + lby*by + lbz*bz);
    cosd_t[i] = cosd * s;

    float proj = bx*lax + by*lay + bz*laz;
    float ox = lax - proj*bx, oy = lay - proj*by, oz = laz - proj*bz;
    float on = sqrtf(ox*ox + oy*oy + oz*oz);
    float dn = fmaxf(on, 1e-12f);
    ox /= dn; oy /= dn; oz /= dn;
    float cosa = 1.0f - (ox*ax + oy*ay + oz*az);
    cosa_t[i] = cosa * s;

    float offacc = 0.f;
    #pragma unroll
    for (int k = 0; k < NBIN; ++k) {
        float x = goff[i*NBIN+k];
        float y = olab[i*NBIN+k];
        float bce = y * softplusf(-x) + (1.0f - y) * softplusf(x);
        offacc += binw[k] * bce;
    }
    off_t[i] = (offacc / (float)NBIN) * s;

    float h = bsh[i];
    bce_t[i] = s * softplusf(-h) + (1.0f - s) * softplusf(h);
}

// ---------------------------------------------------------------------------
// Kernel 2: pred -> GT min distance (adds loss). One wave32 per 16 pred rows.
// ---------------------------------------------------------------------------
__global__ __launch_bounds__(32) void pcf_p2g(
    const float* __restrict__ Apred, const float* __restrict__ Gpos,
    const float* __restrict__ Gsym,
    const float* __restrict__ n_pred, const float* __restrict__ n_pos,
    const float* __restrict__ n_sym, float* __restrict__ best)
{
    int wid = blockIdx.x;
    int b   = wid / NT16;
    int n0  = (wid % NT16) * 16;
    int lane = threadIdx.x;
    int grp = lane >> 4;
    int r   = lane & 15;

    size_t arow = (size_t)(b*NN + n0 + r) * 16;
    float4 av0 = *(const float4*)(Apred + arow + 0);
    float4 av1 = *(const float4*)(Apred + arow + 4);
    float4 av2 = *(const float4*)(Apred + arow + 8);
    float4 av3 = *(const float4*)(Apred + arow + 12);
    v2f a0 = pick2(av0, grp), a1 = pick2(av1, grp);
    v2f a2 = pick2(av2, grp), a3 = pick2(av3, grp);

    float arown[8];
    #pragma unroll
    for (int k = 0; k < 8; ++k) arown[k] = n_pred[b*NN + n0 + k + 8*grp];

    float runmin[8];
    #pragma unroll
    for (int k = 0; k < 8; ++k) runmin[k] = 3.4e38f;

    for (int m0 = 0; m0 < NN; m0 += 16) {
        size_t grow = (size_t)(b*NN + m0 + r) * 16;
        float4 p0 = *(const float4*)(Gpos + grow + 0);
        float4 p1 = *(const float4*)(Gpos + grow + 4);
        float4 p2 = *(const float4*)(Gpos + grow + 8);
        float4 p3 = *(const float4*)(Gpos + grow + 12);
        float4 s0 = *(const float4*)(Gsym + grow + 0);
        float4 s1 = *(const float4*)(Gsym + grow + 4);
        float4 s2 = *(const float4*)(Gsym + grow + 8);
        float4 s3 = *(const float4*)(Gsym + grow + 12);

        v8f acc1 = wmma_k16(a0,a1,a2,a3, pick2(p0,grp),pick2(p1,grp),pick2(p2,grp),pick2(p3,grp));
        v8f acc2 = wmma_k16(a0,a1,a2,a3, pick2(s0,grp),pick2(s1,grp),pick2(s2,grp),pick2(s3,grp));

        float g1 = n_pos[b*NN + m0 + r];
        float g2 = n_sym[b*NN + m0 + r];
        #pragma unroll
        for (int k = 0; k < 8; ++k) {
            float d1 = fmaxf(arown[k] + g1 - 2.0f*acc1[k], 0.0f);
            float d2 = fmaxf(arown[k] + g2 - 2.0f*acc2[k], 0.0f);
            runmin[k] = fminf(runmin[k], fminf(d1, d2));
        }
    }

    #pragma unroll
    for (int k = 0; k < 8; ++k) {
        float v = runmin[k];
        #pragma unroll
        for (int m = 1; m < 16; m <<= 1)
            v = fminf(v, __shfl_xor(v, m, 32));
        if (r == 0) best[b*NN + n0 + k + 8*grp] = v;
    }
}

// ---------------------------------------------------------------------------
// Kernel 3: GT -> pred min + argmin (gt2pred loss). One wave32 per 16 GT rows.
// ---------------------------------------------------------------------------
__global__ __launch_bounds__(32) void pcf_g2p(
    const float* __restrict__ Ggt, const float* __restrict__ Gsgt,
    const float* __restrict__ Apred,
    const float* __restrict__ n_gt, const float* __restrict__ n_sgt,
    const float* __restrict__ n_pred,
    const float* __restrict__ bsp, const float* __restrict__ succ,
    float* __restrict__ g2p_t)
{
    int wid = blockIdx.x;
    int b   = wid / NT16;
    int n0  = (wid % NT16) * 16;
    int lane = threadIdx.x;
    int grp = lane >> 4;
    int r   = lane & 15;

    size_t arow = (size_t)(b*NN + n0 + r) * 16;
    float4 g0 = *(const float4*)(Ggt + arow + 0);
    float4 g1 = *(const float4*)(Ggt + arow + 4);
    float4 g2 = *(const float4*)(Ggt + arow + 8);
    float4 g3 = *(const float4*)(Ggt + arow + 12);
    float4 h0 = *(const float4*)(Gsgt + arow + 0);
    float4 h1 = *(const float4*)(Gsgt + arow + 4);
    float4 h2 = *(const float4*)(Gsgt + arow + 8);
    float4 h3 = *(const float4*)(Gsgt + arow + 12);
    v2f ag0 = pick2(g0,grp), ag1 = pick2(g1,grp), ag2 = pick2(g2,grp), ag3 = pick2(g3,grp);
    v2f as0 = pick2(h0,grp), as1 = pick2(h1,grp), as2 = pick2(h2,grp), as3 = pick2(h3,grp);

    float ng[8], ns[8];
    #pragma unroll
    for (int k = 0; k < 8; ++k) {
        ng[k] = n_gt [b*NN + n0 + k + 8*grp];
        ns[k] = n_sgt[b*NN + n0 + k + 8*grp];
    }

    float m1[8], m2[8]; int i1[8], i2[8];
    #pragma unroll
    for (int k = 0; k < 8; ++k) { m1[k]=3.4e38f; m2[k]=3.4e38f; i1[k]=0; i2[k]=0; }

    for (int m0 = 0; m0 < NN; m0 += 16) {
        size_t prow = (size_t)(b*NN + m0 + r) * 16;
        float4 p0 = *(const float4*)(Apred + prow + 0);
        float4 p1 = *(const float4*)(Apred + prow + 4);
        float4 p2 = *(const float4*)(Apred + prow + 8);
        float4 p3 = *(const float4*)(Apred + prow + 12);
        v2f b0 = pick2(p0,grp), b1 = pick2(p1,grp), b2 = pick2(p2,grp), b3 = pick2(p3,grp);

        v8f acc1 = wmma_k16(ag0,ag1,ag2,ag3, b0,b1,b2,b3);
        v8f acc2 = wmma_k16(as0,as1,as2,as3, b0,b1,b2,b3);

        float np = n_pred[b*NN + m0 + r];
        int col = m0 + r;
        #pragma unroll
        for (int k = 0; k < 8; ++k) {
            float d1 = fmaxf(ng[k] + np - 2.0f*acc1[k], 0.0f);
            if (d1 < m1[k]) { m1[k] = d1; i1[k] = col; }
            float d2 = fmaxf(ns[k] + np - 2.0f*acc2[k], 0.0f);
            if (d2 < m2[k]) { m2[k] = d2; i2[k] = col; }
        }
    }

    #pragma unroll
    for (int k = 0; k < 8; ++k) {
        float v1 = m1[k]; int j1 = i1[k];
        float v2 = m2[k]; int j2 = i2[k];
        #pragma unroll
        for (int m = 1; m < 16; m <<= 1) {
            float ov1 = __shfl_xor(v1, m, 32); int oj1 = __shfl_xor(j1, m, 32);
            if (ov1 < v1 || (ov1 == v1 && oj1 < j1)) { v1 = ov1; j1 = oj1; }
            float ov2 = __shfl_xor(v2, m, 32); int oj2 = __shfl_xor(j2, m, 32);
            if (ov2 < v2 || (ov2 == v2 && oj2 < j2)) { v2 = ov2; j2 = oj2; }
        }
        if (r == 0) {
            int n = b*NN + n0 + k + 8*grp;
            float mm = fminf(v1, v2);
            int idx = (v2 < v1) ? j2 : j1;
            float conf = bsp[b*NN + idx];
            g2p_t[n] = conf * mm * succ[n];
        }
    }
}

// ---------------------------------------------------------------------------
// Kernel 4: deterministic single-block final reduction -> total loss scalar.
// ---------------------------------------------------------------------------
__device__ __forceinline__ float block_reduce_sum(float v, float* red) {
    int t = threadIdx.x;
    red[t] = v;
    __syncthreads();
    for (int o = 128; o > 0; o >>= 1) {
        if (t < o) red[t] += red[t + o];
        __syncthreads();
    }
    float r = red[0];
    __syncthreads();
    return r;
}

__global__ __launch_bounds__(256) void pcf_reduce(
    const float* __restrict__ succ, const float* __restrict__ bsp,
    const float* __restrict__ best, const float* __restrict__ g2p_t,
    const float* __restrict__ cosd_t, const float* __restrict__ cosa_t,
    const float* __restrict__ off_t,  const float* __restrict__ bce_t,
    float* __restrict__ out)
{
    __shared__ float red[256];
    __shared__ float piv[BB], gate[BB];
    int t = threadIdx.x;

    for (int b = 0; b < BB; ++b) {
        float s = 0.f;
        for (int n = t; n < NN; n += 256) s += succ[b*NN + n];
        float tot = block_reduce_sum(s, red);
        if (t == 0) { piv[b] = fmaxf(tot, 1.0f); gate[b] = fminf(tot, 1.0f); }
        __syncthreads();
    }

    float dirL = 0.f, appL = 0.f, offL = 0.f, g2pL = 0.f;
    for (int b = 0; b < BB; ++b) {
        float s1=0.f, s2=0.f, s3=0.f, s4=0.f;
        for (int n = t; n < NN; n += 256) {
            int idx = b*NN + n;
            s1 += cosd_t[idx]; s2 += cosa_t[idx];
            s3 += off_t[idx];  s4 += g2p_t[idx];
        }
        float r1 = block_reduce_sum(s1, red);
        float r2 = block_reduce_sum(s2, red);
        float r3 = block_reduce_sum(s3, red);
        float r4 = block_reduce_sum(s4, red);
        if (t == 0) {
            dirL += r1 / piv[b]; appL += r2 / piv[b];
            offL += r3 / piv[b]; g2pL += r4 / piv[b];
        }
    }

    float sa = 0.f, sb = 0.f;
    for (int idx = t; idx < BN; idx += 256) {
        int b = idx / NN;
        sa += bsp[idx] * gate[b] * sqrtf(best[idx]);
        sb += bce_t[idx];
    }
    float ra = block_reduce_sum(sa, red);
    float rb = block_reduce_sum(sb, red);

    if (t == 0) {
        float adds = ra / (float)BN;
        float bce  = rb / (float)BN;
        float total = dirL/(float)BB + bce + offL/(float)BB + appL/(float)BB
                    + 10.0f * adds + g2pL/(float)BB;
        out[0] = total;
    }
}

// ---------------------------------------------------------------------------
extern "C" void kernel_launch(void* const* d_in, const int* in_sizes, int n_in,
                              void* d_out, int out_size, void* d_ws, size_t ws_size,
                              hipStream_t stream) {
    const float* gdir  = (const float*)d_in[0];
    const float* adir  = (const float*)d_in[1];
    const float* goff  = (const float*)d_in[2];
    const float* ppts  = (const float*)d_in[3];
    const float* bsp   = (const float*)d_in[4];
    const float* bsh   = (const float*)d_in[5];
    const float* dlab  = (const float*)d_in[6];
    const float* olab  = (const float*)d_in[7];
    const float* succ  = (const float*)d_in[8];
    const float* alab  = (const float*)d_in[9];
    const float* binv  = (const float*)d_in[10];
    const float* binw  = (const float*)d_in[11];
    const float* cpts  = (const float*)d_in[12];
    const float* scpts = (const float*)d_in[13];
    float* out = (float*)d_out;

    // Workspace partition (~2.85 MB of floats)
    float* w = (float*)d_ws;
    float* Apred = w;  w += (size_t)BN * 16;
    float* Gpos  = w;  w += (size_t)BN * 16;
    float* Gsym  = w;  w += (size_t)BN * 16;
    float* Ggt   = w;  w += (size_t)BN * 16;
    float* Gsgt  = w;  w += (size_t)BN * 16;
    float* n_pred = w; w += BN;
    float* n_pos  = w; w += BN;
    float* n_sym  = w; w += BN;
    float* n_gt   = w; w += BN;
    float* n_sgt  = w; w += BN;
    float* bestb  = w; w += BN;
    float* g2p_t  = w; w += BN;
    float* cosd_t = w; w += BN;
    float* cosa_t = w; w += BN;
    float* off_t  = w; w += BN;
    float* bce_t  = w; w += BN;

    pcf_setup<<<(BN + 255) / 256, 256, 0, stream>>>(
        gdir, adir, goff, ppts, bsp, bsh, dlab, olab, succ, alab, binv, binw,
        cpts, scpts,
        Apred, Gpos, Gsym, Ggt, Gsgt,
        n_pred, n_pos, n_sym, n_gt, n_sgt,
        cosd_t, cosa_t, off_t, bce_t);

    pcf_p2g<<<BB * NT16, 32, 0, stream>>>(
        Apred, Gpos, Gsym, n_pred, n_pos, n_sym, bestb);

    pcf_g2p<<<BB * NT16, 32, 0, stream>>>(
        Ggt, Gsgt, Apred, n_gt, n_sgt, n_pred, bsp, succ, g2p_t);

    pcf_reduce<<<1, 256, 0, stream>>>(
        succ, bsp, bestb, g2p_t, cosd_t, cosa_t, off_t, bce_t, out);
}